// ExpertChoiceRouter_87978110091811
// MI455X (gfx1250) — compile-verified
//
#include <hip/hip_runtime.h>

#define BB 4
#define SS 8192
#define HH 4096
#define EE 64
#define NN (BB * SS)   /* 32768 tokens */
#define KTOP 512       /* N * capacity_factor / E */

typedef float v2f __attribute__((ext_vector_type(2)));
typedef float v8f __attribute__((ext_vector_type(8)));

__device__ __forceinline__ unsigned mono_key(float f) {
  // order-preserving map float -> uint (larger key == larger float)
  unsigned b = __float_as_uint(f);
  return (b & 0x80000000u) ? ~b : (b | 0x80000000u);
}

// ---------------------------------------------------------------------------
// Kernel 1: logits[N,E] = x[N,H] @ W[H,E], fp32 WMMA 16x16x4.
// grid.x = NN/64, block = 128 threads (4 waves). Wave w computes rows
// [64*blk + 16w, +16) x all 64 expert columns (4 column tiles of 16).
// W tile staged TRANSPOSED in LDS so each B fragment is one aligned b64 load.
// ---------------------------------------------------------------------------
__global__ __launch_bounds__(128)
void gemm_logits(const float* __restrict__ x, const float* __restrict__ W,
                 float* __restrict__ logits) {
  __shared__ float As[64][68];    // [row][k]  pitch 68: even, conflict-free b64 reads
  __shared__ float WsT[64][70];   // [e][k]    pitch 70: even (aligned b64), bank-spread

  const int tid  = threadIdx.x;
  const int wave = tid >> 5;
  const int lane = tid & 31;
  const int half = lane >> 4;     // 0: lanes 0-15, 1: lanes 16-31
  const int lr   = lane & 15;
  const int m0   = blockIdx.x * 64;

  v8f acc0 = {}, acc1 = {}, acc2 = {}, acc3 = {};

  for (int kk = 0; kk < HH; kk += 64) {
    // Cooperative float4 loads: 64x64 A tile (row-major) and 64x64 W tile
    // (transposed on the fly into WsT[e][k]).
    for (int i = tid; i < 64 * 16; i += 128) {
      const int r  = i >> 4;            // k-row within tile
      const int c4 = (i & 15) << 2;     // column group of 4
      const float4 av = *(const float4*)&x[(size_t)(m0 + r) * HH + kk + c4];
      As[r][c4 + 0] = av.x; As[r][c4 + 1] = av.y;
      As[r][c4 + 2] = av.z; As[r][c4 + 3] = av.w;
      const float4 wv = *(const float4*)&W[(size_t)(kk + r) * EE + c4];
      WsT[c4 + 0][r] = wv.x; WsT[c4 + 1][r] = wv.y;
      WsT[c4 + 2][r] = wv.z; WsT[c4 + 3][r] = wv.w;
    }
    // Prefetch next K-slab of x (emits global_prefetch_b8).
    if (kk + 64 < HH && tid < 64) {
      __builtin_prefetch(&x[(size_t)(m0 + tid) * HH + kk + 64], 0, 1);
    }
    __syncthreads();

#pragma unroll 4
    for (int k2 = 0; k2 < 64; k2 += 4) {
      const int kh = k2 + 2 * half;   // even -> 8B-aligned LDS b64 loads
      // A 16x4 fp32 frag: lanes 0-15 -> {K0,K1} ; lanes 16-31 -> {K2,K3}
      const v2f a  = *(const v2f*)&As[wave * 16 + lr][kh];
      // B 4x16 fp32 frags: contiguous {K,K+1} pair per lane, one b64 each
      const v2f b0 = *(const v2f*)&WsT[ 0 + lr][kh];
      const v2f b1 = *(const v2f*)&WsT[16 + lr][kh];
      const v2f b2 = *(const v2f*)&WsT[32 + lr][kh];
      const v2f b3 = *(const v2f*)&WsT[48 + lr][kh];

      acc0 = __builtin_amdgcn_wmma_f32_16x16x4_f32(false, a, false, b0, (short)0, acc0, false, false);
      acc1 = __builtin_amdgcn_wmma_f32_16x16x4_f32(false, a, false, b1, (short)0, acc1, false, false);
      acc2 = __builtin_amdgcn_wmma_f32_16x16x4_f32(false, a, false, b2, (short)0, acc2, false, false);
      acc3 = __builtin_amdgcn_wmma_f32_16x16x4_f32(false, a, false, b3, (short)0, acc3, false, false);
    }
    __syncthreads();
  }

  // C/D layout: VGPR v -> M = v (lanes 0-15) / v+8 (lanes 16-31), N = lr.
  const int row0 = m0 + wave * 16 + half * 8;
#pragma unroll
  for (int v = 0; v < 8; ++v) {
    const size_t r = (size_t)(row0 + v) * EE;
    logits[r +  0 + lr] = acc0[v];
    logits[r + 16 + lr] = acc1[v];
    logits[r + 32 + lr] = acc2[v];
    logits[r + 48 + lr] = acc3[v];
  }
}

// ---------------------------------------------------------------------------
// Kernel 2a: transpose logits [N,E] -> monotonic keys [E,N] (coalesced).
// ---------------------------------------------------------------------------
__global__ __launch_bounds__(256)
void transpose_keys(const float* __restrict__ logits, unsigned* __restrict__ keysT) {
  __shared__ float tile[64][65];
  const int t0 = blockIdx.x * 64;
  for (int i = threadIdx.x; i < 64 * 64; i += 256) {
    const int r = i >> 6, c = i & 63;
    tile[r][c] = logits[(size_t)(t0 + r) * EE + c];
  }
  __syncthreads();
  for (int i = threadIdx.x; i < 64 * 64; i += 256) {
    const int e = i >> 6, r = i & 63;
    keysT[(size_t)e * NN + t0 + r] = mono_key(tile[r][e]);
  }
}

// ---------------------------------------------------------------------------
// Kernel 2b: per-expert k-th largest key via 32-step bisection (L2-resident).
// One block (256 threads) per expert.
// ---------------------------------------------------------------------------
__global__ __launch_bounds__(256)
void expert_threshold(const unsigned* __restrict__ keysT, unsigned* __restrict__ thrKey) {
  const int e = blockIdx.x;
  const unsigned* col = keysT + (size_t)e * NN;
  __shared__ unsigned red[256];

  unsigned lo = 0u, hi = 0xFFFFFFFFu;  // invariant: count(key >= lo) >= KTOP
  for (int it = 0; it < 32 && lo < hi; ++it) {
    const unsigned mid = lo + ((hi - lo) >> 1) + 1u;  // candidate in (lo, hi]
    unsigned c = 0;
    for (int t = threadIdx.x; t < NN; t += 256) c += (col[t] >= mid) ? 1u : 0u;
    red[threadIdx.x] = c;
    __syncthreads();
    for (int s = 128; s > 0; s >>= 1) {
      if (threadIdx.x < s) red[threadIdx.x] += red[threadIdx.x + s];
      __syncthreads();
    }
    const unsigned total = red[0];
    __syncthreads();
    if (total >= KTOP) lo = mid; else hi = mid - 1u;
  }
  if (threadIdx.x == 0) thrKey[e] = lo;  // = KTOP-th largest key in column e
}

// ---------------------------------------------------------------------------
// Kernel 3: per-token assignment + fallback argmax + weight.
// ---------------------------------------------------------------------------
__global__ __launch_bounds__(256)
void assign_tokens(const float* __restrict__ logits, const unsigned* __restrict__ thrKey,
                   float* __restrict__ rout, float* __restrict__ eidx) {
  __shared__ unsigned thr_s[EE];
  if (threadIdx.x < EE) thr_s[threadIdx.x] = thrKey[threadIdx.x];
  __syncthreads();

  const int t = blockIdx.x * blockDim.x + threadIdx.x;
  if (t >= NN) return;
  const float* row = logits + (size_t)t * EE;

  float fbBest = -__builtin_huge_valf(); int fbIdx = 0;   // argmax fallback
  float aBest  = -__builtin_huge_valf(); int aIdx  = EE;  // best claiming expert

#pragma unroll 8
  for (int e = 0; e < EE; ++e) {
    const float v = row[e];
    if (v > fbBest) { fbBest = v; fbIdx = e; }            // first max kept
    if (mono_key(v) >= thr_s[e] && v > aBest) {           // strict > : earliest
      aBest = v; aIdx = e;                                // expert wins ties
    }
  }
  const int fin = (aIdx < EE) ? aIdx : fbIdx;
  rout[t] = row[fin];
  eidx[t] = (float)fin;
}

// ---------------------------------------------------------------------------
extern "C" void kernel_launch(void* const* d_in, const int* in_sizes, int n_in,
                              void* d_out, int out_size, void* d_ws, size_t ws_size,
                              hipStream_t stream) {
  (void)in_sizes; (void)n_in; (void)out_size; (void)ws_size;

  const float* x = (const float*)d_in[0];   // [B,S,H] fp32
  const float* W = (const float*)d_in[1];   // [H,E]   fp32

  // d_out: routing_weights [N] | expert_indices [N] | logits [N,E]
  float* rout   = (float*)d_out;
  float* eidx   = rout + NN;
  float* logits = eidx + NN;

  // d_ws: transposed monotonic keys [E,N] (8 MB) | per-expert thresholds [E]
  unsigned* keysT  = (unsigned*)d_ws;
  unsigned* thrKey = keysT + (size_t)EE * NN;

  gemm_logits     <<<NN / 64,  128, 0, stream>>>(x, W, logits);
  transpose_keys  <<<NN / 64,  256, 0, stream>>>(logits, keysT);
  expert_threshold<<<EE,       256, 0, stream>>>(keysT, thrKey);
  assign_tokens   <<<NN / 256, 256, 0, stream>>>(logits, thrKey, rout, eidx);
}